// GCN_20615843021612
// MI455X (gfx1250) — compile-verified
//
#include <hip/hip_runtime.h>
#include <hip/hip_bf16.h>

typedef float v2f __attribute__((ext_vector_type(2)));
typedef float v8f __attribute__((ext_vector_type(8)));

#define KDIM 128  // IN_C == HID == 128 for every GEMM's K dimension

// ---------------------------------------------------------------------------
// Degree / normalization kernels
// ---------------------------------------------------------------------------
__global__ void gcn_init_deg(float* deg, int n) {
    int i = blockIdx.x * blockDim.x + threadIdx.x;
    if (i < n) deg[i] = 1.0f;  // self-loop contribution
}

__global__ void gcn_deg_scatter(const int* __restrict__ dst, float* deg, int e) {
    int i = blockIdx.x * blockDim.x + threadIdx.x;
    if (i < e) unsafeAtomicAdd(&deg[dst[i]], 1.0f);
}

__global__ void gcn_finalize_dis(float* dis, int n) {
    int i = blockIdx.x * blockDim.x + threadIdx.x;
    if (i < n) dis[i] = __frsqrt_rn(dis[i]);  // deg >= 1 always (self loops)
}

// ---------------------------------------------------------------------------
// Weight transpose: Wt[n*K + k] = W[k*N + n]   (K = 128 always; W is tiny)
// Makes the WMMA B-fragment loads contiguous (one b64 per k-step).
// ---------------------------------------------------------------------------
__global__ void gcn_transpose(const float* __restrict__ W, float* __restrict__ Wt, int N) {
    int idx = blockIdx.x * blockDim.x + threadIdx.x;  // over K*N
    if (idx < KDIM * N) {
        int k = idx / N, n = idx - k * N;
        Wt[n * KDIM + k] = W[idx];
    }
}

// ---------------------------------------------------------------------------
// WMMA fp32 GEMM:  C[M,N] = A[M,128] * Wt[N,128]^T  (+ optional bias)
// One wave computes one 16x16 output tile via 32 chained V_WMMA_F32_16X16X4_F32.
// A-frag layout (32-bit A 16x4): lanes 0-15 -> K={0,1}, lanes 16-31 -> K={2,3}.
// B-frag mirrors A via the transposed weights, so both sides are single b64
// loads with compile-time immediate offsets (N templated).
// C/D layout: VGPR i holds M=i (lanes 0-15) and M=i+8 (lanes 16-31), N=lane&15.
// Requires M % 16 == 0 (M = 50000 = 3125 * 16).
// ---------------------------------------------------------------------------
template <int N>
__global__ void gcn_gemm_wmma(const float* __restrict__ A,
                              const float* __restrict__ Wt,
                              float* __restrict__ C,
                              const float* __restrict__ bias,
                              int totalTiles) {
    const int lane = threadIdx.x & 31;
    const int tile = blockIdx.x * 8 + (threadIdx.x >> 5);
    if (tile >= totalTiles) return;              // whole wave exits: EXEC stays all-1s

    constexpr int ntilesN = N / 16;
    const int tm = tile / ntilesN;
    const int tn = tile - tm * ntilesN;

    const int mLocal = lane & 15;
    const int kHalf  = (lane >> 4) << 1;         // 0 for lanes 0-15, 2 for lanes 16-31
    const int colN   = tn * 16 + mLocal;

    const float* __restrict__ arow = A  + (size_t)(tm * 16 + mLocal) * KDIM + kHalf;
    const float* __restrict__ brow = Wt + (size_t)colN * KDIM + kHalf;

    v8f c = {};
#pragma unroll
    for (int k0 = 0; k0 < KDIM; k0 += 4) {
        v2f a = *(const v2f*)(arow + k0);
        v2f b = *(const v2f*)(brow + k0);
        c = __builtin_amdgcn_wmma_f32_16x16x4_f32(
                /*neg_a=*/false, a, /*neg_b=*/false, b,
                /*c_mod=*/(short)0, c, /*reuse_a=*/false, /*reuse_b=*/false);
    }

    const float badd = bias ? bias[colN] : 0.0f;
    const int mBase = tm * 16 + ((lane >> 4) << 3);
#pragma unroll
    for (int i = 0; i < 8; ++i)
        C[(size_t)(mBase + i) * N + colN] = c[i] + badd;
}

// ---------------------------------------------------------------------------
// Edge aggregation: out[dst] += dis[src]*dis[dst] * h[src]
// One wave per edge; each lane moves 4 consecutive channels (float4).
// Node features (25.6 MB) are fully L2-resident -> L2-atomic bound.
// ---------------------------------------------------------------------------
__global__ void gcn_edge_scatter(const float* __restrict__ h,
                                 const float* __restrict__ dis,
                                 const int* __restrict__ src,
                                 const int* __restrict__ dst,
                                 float* out, int e) {
    const int lane = threadIdx.x & 31;
    const int edge = blockIdx.x * 8 + (threadIdx.x >> 5);
    if (edge >= e) return;
    const int s = src[edge];
    const int d = dst[edge];
    const float norm = dis[s] * dis[d];
    const float4 v = ((const float4*)(h + (size_t)s * 128))[lane];
    float* o = out + (size_t)d * 128 + lane * 4;
    unsafeAtomicAdd(o + 0, v.x * norm);
    unsafeAtomicAdd(o + 1, v.y * norm);
    unsafeAtomicAdd(o + 2, v.z * norm);
    unsafeAtomicAdd(o + 3, v.w * norm);
}

// ---------------------------------------------------------------------------
// Epilogue: act = relu(agg + h*dis^2 (self loop) + b)   [in place over agg]
// ---------------------------------------------------------------------------
__global__ void gcn_self_bias_relu(float* agg, const float* __restrict__ h,
                                   const float* __restrict__ dis,
                                   const float* __restrict__ b) {
    const int node = blockIdx.x;
    const int c = threadIdx.x;                   // 128 threads
    const float d = dis[node];
    const size_t idx = (size_t)node * 128 + c;
    float v = agg[idx] + h[idx] * d * d + b[c];
    agg[idx] = fmaxf(v, 0.0f);
}

// ---------------------------------------------------------------------------
extern "C" void kernel_launch(void* const* d_in, const int* in_sizes, int n_in,
                              void* d_out, int out_size, void* d_ws, size_t ws_size,
                              hipStream_t stream) {
    const float* x  = (const float*)d_in[0];
    const int*   ei = (const int*)d_in[1];
    const float* W1 = (const float*)d_in[2];
    const float* b1 = (const float*)d_in[3];
    const float* W2 = (const float*)d_in[4];
    const float* b2 = (const float*)d_in[5];
    const float* Wh = (const float*)d_in[6];
    const float* bh = (const float*)d_in[7];
    float* out = (float*)d_out;

    const int M = in_sizes[0] / KDIM;     // 50000 nodes (multiple of 16)
    const int E = in_sizes[1] / 2;        // 800000 edges
    const int* src = ei;
    const int* dst = ei + E;

    // Workspace layout: dis[M] | Wt1[128*128] | Wt2[128*128] | Wth[16*128]
    //                   | bufH[M*128] | bufAgg[M*128]
    char* ws = (char*)d_ws;
    float* dis = (float*)ws;
    size_t off = (((size_t)M * 4) + 255) & ~(size_t)255;
    float* Wt1 = (float*)(ws + off);
    float* Wt2 = Wt1 + 128 * KDIM;
    float* Wth = Wt2 + 128 * KDIM;
    float* bufH   = Wth + 16 * KDIM;
    float* bufAgg = bufH + (size_t)M * 128;

    const int T256N = (M + 255) / 256;
    const int tiles128 = (M / 16) * (128 / 16);
    const int tilesCls = (M / 16) * (16 / 16);

    // --- weight transposes (tiny) ---
    gcn_transpose<<<(KDIM * 128 + 255) / 256, 256, 0, stream>>>(W1, Wt1, 128);
    gcn_transpose<<<(KDIM * 128 + 255) / 256, 256, 0, stream>>>(W2, Wt2, 128);
    gcn_transpose<<<(KDIM * 16  + 255) / 256, 256, 0, stream>>>(Wh, Wth, 16);

    // --- symmetric normalization D^{-1/2} ---
    gcn_init_deg<<<T256N, 256, 0, stream>>>(dis, M);
    gcn_deg_scatter<<<(E + 255) / 256, 256, 0, stream>>>(dst, dis, E);
    gcn_finalize_dis<<<T256N, 256, 0, stream>>>(dis, M);

    // --- layer 1 ---
    gcn_gemm_wmma<128><<<(tiles128 + 7) / 8, 256, 0, stream>>>(x, Wt1, bufH, nullptr, tiles128);
    hipMemsetAsync(bufAgg, 0, (size_t)M * 128 * sizeof(float), stream);
    gcn_edge_scatter<<<(E + 7) / 8, 256, 0, stream>>>(bufH, dis, src, dst, bufAgg, E);
    gcn_self_bias_relu<<<M, 128, 0, stream>>>(bufAgg, bufH, dis, b1);

    // --- layer 2 ---
    gcn_gemm_wmma<128><<<(tiles128 + 7) / 8, 256, 0, stream>>>(bufAgg, Wt2, bufH, nullptr, tiles128);
    hipMemsetAsync(bufAgg, 0, (size_t)M * 128 * sizeof(float), stream);
    gcn_edge_scatter<<<(E + 7) / 8, 256, 0, stream>>>(bufH, dis, src, dst, bufAgg, E);
    gcn_self_bias_relu<<<M, 128, 0, stream>>>(bufAgg, bufH, dis, b2);

    // --- classification head: [M,128] x [128,16] + bh -> d_out ---
    gcn_gemm_wmma<16><<<(tilesCls + 7) / 8, 256, 0, stream>>>(bufAgg, Wth, out, bh, tilesCls);
}